// GPSConv_72679436583234
// MI455X (gfx1250) — compile-verified
//
#include <hip/hip_runtime.h>

// ---------------------------------------------------------------------------
// GPSConv forward for MI455X (gfx1250, wave32)
// fp32 WMMA 16x16x4 GEMMs + Tensor Data Mover (TDM) LDS tile staging
// B=32, L=512, C=256, H=8, d=32, N=16384
// ---------------------------------------------------------------------------

#define B_ 32
#define L_ 512
#define C_ 256
#define H_ 8
#define D_ 32
#define N_ (B_ * L_)

typedef float v2f __attribute__((ext_vector_type(2)));
typedef float v8f __attribute__((ext_vector_type(8)));
typedef unsigned int u32;
typedef unsigned long long u64;

__device__ __forceinline__ v8f wmma4(v2f a, v2f b, v8f c) {
  // D = A(16x4 f32) * B(4x16 f32) + C(16x16 f32)
  return __builtin_amdgcn_wmma_f32_16x16x4_f32(false, a, false, b, (short)0, c,
                                               false, false);
}

// ---------------------------------------------------------------------------
// Tile staging: TDM (tensor_load_to_lds) when available, manual loads else.
// All staged tiles are [rows][36] f32: 32 data cols + 4 pad cols, which the
// TDM pad feature reproduces (pad 4 DWORDs after every 32 DWORDs stored).
// ---------------------------------------------------------------------------
#if __has_builtin(__builtin_amdgcn_tensor_load_to_lds)
#define HAVE_TDM 1
#else
#define HAVE_TDM 0
#endif

#if HAVE_TDM
typedef u32 v4u __attribute__((ext_vector_type(4)));
typedef int v4i __attribute__((ext_vector_type(4)));
typedef int v8i __attribute__((ext_vector_type(8)));

// 2D tile DMA: rows x 32 f32 elements, row stride `stride` elements,
// written to LDS at byte offset lds_off with +16B padding per row.
__device__ __forceinline__ void tdm_load_tile(u32 lds_off, const void* g,
                                              u32 rows, u32 stride) {
  const u64 ga = (u64)g;
  v4u g0;
  g0.x = 1u;                                        // count=1, user descriptor
  g0.y = lds_off;                                   // lds_addr (bytes)
  g0.z = (u32)ga;                                   // global_addr[31:0]
  g0.w = ((u32)(ga >> 32) & 0x01FFFFFFu) | (2u << 30);  // addr[56:32], type=2
  v8i g1;
  // data_size=2 (4B) | pad_enable | pad_interval=4 (32 DW) | pad_amount=3 (4 DW)
  g1[0] = (int)((2u << 16) | (1u << 20) | (4u << 22) | (3u << 25));
  g1[1] = (int)((32u & 0xFFFFu) << 16);             // tensor_dim0 = 32 (lo16)
  g1[2] = (int)((32u >> 16) | ((rows & 0xFFFFu) << 16));  // dim0 hi | dim1 lo
  g1[3] = (int)((rows >> 16) | (32u << 16));        // dim1 hi | tile_dim0 = 32
  g1[4] = (int)rows;                                // tile_dim1, tile_dim2=0
  g1[5] = (int)stride;                              // tensor_dim0_stride[31:0]
  g1[6] = 0;                                        // stride hi | dim1_stride lo
  g1[7] = 0;
  const v4i z4 = {0, 0, 0, 0};
#if __clang_major__ >= 23
  const v8i z8 = {0, 0, 0, 0, 0, 0, 0, 0};
  __builtin_amdgcn_tensor_load_to_lds(g0, g1, z4, z4, z8, 0);
#else
  __builtin_amdgcn_tensor_load_to_lds(g0, g1, z4, z4, 0);
#endif
}

__device__ __forceinline__ void stage_tile(float (*dst)[36], const float* src,
                                           u32 rows, u32 stride, int t) {
  (void)t;
  tdm_load_tile((u32)(u64)&dst[0][0], src, rows, stride);
}
#define STAGE_WAIT(n)                       \
  do {                                      \
    __builtin_amdgcn_s_wait_tensorcnt(n);   \
    asm volatile("" ::: "memory");          \
  } while (0)
#else
__device__ __forceinline__ void stage_tile(float (*dst)[36], const float* src,
                                           u32 rows, u32 stride, int t) {
  for (u32 r = (u32)(t >> 1); r < rows; r += 16) {
    const int c = (t & 1) * 16;
    const float* s = src + (u64)r * stride + c;
    *(float4*)&dst[r][c + 0] = *(const float4*)(s + 0);
    *(float4*)&dst[r][c + 4] = *(const float4*)(s + 4);
    *(float4*)&dst[r][c + 8] = *(const float4*)(s + 8);
    *(float4*)&dst[r][c + 12] = *(const float4*)(s + 12);
  }
}
#define STAGE_WAIT(n) __syncthreads()
#endif

// ---------------------------------------------------------------------------
// Generic WMMA GEMM: OUT[M,Nc] = epi(X[M,K] @ W[K,Nc] + bias)
// One wave per 16(M) x 32(N) tile. A and B K-chunks double-buffered in LDS
// via TDM; inner loop is pure DS reads + v_wmma_f32_16x16x4_f32.
// mode 0: none   1: relu   2: gated residual: g*(res+v) + (1-g)*x0
// ---------------------------------------------------------------------------
__global__ __launch_bounds__(32) void wmma_gemm_kernel(
    const float* __restrict__ X, const float* __restrict__ W,
    const float* __restrict__ bias, float* __restrict__ OUT, int M, int K,
    int Nc, int mode, const float* __restrict__ res,
    const float* __restrict__ x0, const float* __restrict__ gate) {
  const int m0 = blockIdx.x * 16;
  const int n0 = blockIdx.y * 32;
  const int t = threadIdx.x;
  const int lane = t & 15;
  const int half = t >> 4;

  __shared__ __align__(16) float As[2][16][36];
  __shared__ __align__(16) float Bs[2][32][36];

  v8f acc0 = {0.f, 0.f, 0.f, 0.f, 0.f, 0.f, 0.f, 0.f};
  v8f acc1 = {0.f, 0.f, 0.f, 0.f, 0.f, 0.f, 0.f, 0.f};

  const float* Abase = X + (u64)m0 * K;
  const float* Bbase = W + n0;

  stage_tile(As[0], Abase, 16, (u32)K, t);
  stage_tile(Bs[0], Bbase, 32, (u32)Nc, t);

  int cur = 0;
  for (int kc = 0; kc < K; kc += 32, cur ^= 1) {
    if (kc + 32 < K) {
      stage_tile(As[cur ^ 1], Abase + kc + 32, 16, (u32)K, t);
      stage_tile(Bs[cur ^ 1], Bbase + (u64)(kc + 32) * Nc, 32, (u32)Nc, t);
      STAGE_WAIT(2);  // previous pair complete (in-order), next pair in flight
    } else {
      STAGE_WAIT(0);
    }
#pragma unroll
    for (int kk = 0; kk < 32; kk += 4) {
      v2f a, b0, b1;
      a.x = As[cur][lane][kk + 2 * half];
      a.y = As[cur][lane][kk + 2 * half + 1];
      b0.x = Bs[cur][kk + 2 * half][lane];
      b0.y = Bs[cur][kk + 2 * half + 1][lane];
      b1.x = Bs[cur][kk + 2 * half][16 + lane];
      b1.y = Bs[cur][kk + 2 * half + 1][16 + lane];
      acc0 = wmma4(a, b0, acc0);
      acc1 = wmma4(a, b1, acc1);
    }
  }

#pragma unroll
  for (int j = 0; j < 8; ++j) {
    const int m = m0 + j + 8 * half;
    const int c0 = n0 + lane;
    const int c1 = n0 + 16 + lane;
    float v0 = acc0[j] + bias[c0];
    float v1 = acc1[j] + bias[c1];
    if (mode == 1) {
      v0 = fmaxf(v0, 0.f);
      v1 = fmaxf(v1, 0.f);
    } else if (mode == 2) {
      const float g = gate[m / L_];
      const float r0 = res[(u64)m * Nc + c0];
      const float r1 = res[(u64)m * Nc + c1];
      const float xa = x0[(u64)m * Nc + c0];
      const float xb = x0[(u64)m * Nc + c1];
      v0 = g * (r0 + v0) + (1.f - g) * xa;
      v1 = g * (r1 + v1) + (1.f - g) * xb;
    }
    OUT[(u64)m * Nc + c0] = v0;
    OUT[(u64)m * Nc + c1] = v1;
  }
}

// ---------------------------------------------------------------------------
// scores[i] = hidden[i,:] . w2 + b2      (one wave per row)
// ---------------------------------------------------------------------------
__global__ __launch_bounds__(32) void score_reduce_kernel(
    const float* __restrict__ hidden, const float* __restrict__ w2,
    const float* __restrict__ b2, float* __restrict__ scores) {
  const int i = blockIdx.x;
  const int t = threadIdx.x;
  const float* hrow = hidden + (u64)i * C_;
  float s = 0.f;
#pragma unroll
  for (int c = 0; c < C_ / 32; ++c) s += hrow[t + c * 32] * w2[t + c * 32];
#pragma unroll
  for (int off = 16; off > 0; off >>= 1) s += __shfl_down(s, off, 32);
  if (t == 0) scores[i] = s + b2[0];
}

// ---------------------------------------------------------------------------
// Per-batch: log-softmax + gumbel, rank-based top-k (matches jax.lax.top_k
// tie-breaking: lowest index wins), refined mask, tidx compaction map.
// ---------------------------------------------------------------------------
__global__ __launch_bounds__(512) void select_kernel(
    const float* __restrict__ scores, const float* __restrict__ ratio,
    const float* __restrict__ gumbel, const int* __restrict__ kmax_p,
    int* __restrict__ tidx, int* __restrict__ kper_out) {
  __shared__ float buf[L_];
  __shared__ float z[L_];
  __shared__ unsigned char selected[L_];
  __shared__ unsigned char refined[L_];

  const int b = blockIdx.x;
  const int t = threadIdx.x;
  const int kmax = *kmax_p;
  int kper = (int)ceilf(ratio[b] * (float)L_);
  kper = kper < 1 ? 1 : (kper > L_ ? L_ : kper);

  const float sc = scores[b * L_ + t];
  buf[t] = sc;
  __syncthreads();
  for (int s = L_ / 2; s > 0; s >>= 1) {
    if (t < s) buf[t] = fmaxf(buf[t], buf[t + s]);
    __syncthreads();
  }
  const float mx = buf[0];
  __syncthreads();
  buf[t] = __expf(sc - mx);
  __syncthreads();
  for (int s = L_ / 2; s > 0; s >>= 1) {
    if (t < s) buf[t] += buf[t + s];
    __syncthreads();
  }
  const float lse = __logf(buf[0]);
  // gumbel logits (TAU=1; softmax is rank-monotone, skip it)
  z[t] = (sc - mx - lse) + gumbel[b * L_ + t];
  __syncthreads();

  const float zi = z[t];
  int r = 0;
  for (int j = 0; j < L_; ++j) {
    const float zj = z[j];
    r += (zj > zi) || (zj == zi && j < t);
  }
  selected[t] = (r < kmax) ? 1 : 0;
  __syncthreads();

  int sb = 0;
  for (int j = 0; j < t; ++j) sb += selected[j];
  const int isref = selected[t] && (sb < kper);
  refined[t] = (unsigned char)isref;
  if (isref) tidx[b * L_ + sb] = t;
  __syncthreads();

  int zr = 0;
  for (int j = 0; j < t; ++j) zr += (refined[j] == 0);
  if (!isref && (kper + zr) < kmax) tidx[b * L_ + kper + zr] = t;

  if (t == 0) kper_out[b] = kper;
}

// ---------------------------------------------------------------------------
// h_compact[b,j,:] = (j < kper) ? x[b, tidx[b,j], :] : 0
// ---------------------------------------------------------------------------
__global__ __launch_bounds__(256) void gather_kernel(
    const float* __restrict__ x, const int* __restrict__ tidx,
    const int* __restrict__ kper, float* __restrict__ hc) {
  const int row = blockIdx.x;  // b*512 + j
  const int b = row >> 9, j = row & 511;
  const int c = threadIdx.x;
  float v = 0.f;
  if (j < kper[b]) v = x[(u64)(b * L_ + tidx[row]) * C_ + c];
  hc[(u64)row * C_ + c] = v;
}

// ---------------------------------------------------------------------------
// Attention: one wave per (qtile, head, batch). S panel (16 x 512) in LDS.
// QK^T and P*V via WMMA f32 16x16x4; K/V tiles double-buffered via TDM.
// ---------------------------------------------------------------------------
__global__ __launch_bounds__(32) void attn_kernel(
    const float* __restrict__ qkv, const int* __restrict__ kper_p,
    const int* __restrict__ kmax_p, float* __restrict__ ctx) {
  const int qt = blockIdx.x;  // 0..31
  const int h = blockIdx.y;   // 0..7
  const int b = blockIdx.z;   // 0..31
  const int kmax = *kmax_p;
  const int q0 = qt * 16;
  if (q0 >= kmax) return;  // uniform branch
  const int kper = kper_p[b];
  const int nkt = (kper + 15) >> 4;

  const int t = threadIdx.x;
  const int lane = t & 15;
  const int half = t >> 4;

  __shared__ __align__(16) float Qs[16][36];
  __shared__ __align__(16) float Ks[2][16][36];
  __shared__ __align__(16) float Vs[2][16][36];
  __shared__ __align__(16) float Sp[16][516];  // padded -> conflict-free cols

  const float* qbase = qkv + (u64)(b * L_) * (3 * C_) + h * D_;
  const u32 ld = 3 * C_;

  // stage Q tile + first K tile
  stage_tile(Qs, qbase + (u64)q0 * ld, 16, ld, t);
  stage_tile(Ks[0], qbase + C_, 16, ld, t);

  const float scale = 0.17677669529663687f;  // 1/sqrt(32)

  // ---- phase 1: S = Q K^T * scale, masked, to LDS panel ----
  for (int kt = 0; kt < nkt; ++kt) {
    if (kt + 1 < nkt) {
      stage_tile(Ks[(kt + 1) & 1], qbase + C_ + (u64)((kt + 1) * 16) * ld, 16,
                 ld, t);
      STAGE_WAIT(1);  // Q and K_kt complete, K_{kt+1} in flight
    } else {
      STAGE_WAIT(0);
    }
    v8f accS = {0.f, 0.f, 0.f, 0.f, 0.f, 0.f, 0.f, 0.f};
    float(*K_)[36] = Ks[kt & 1];
#pragma unroll
    for (int kk = 0; kk < 32; kk += 4) {
      v2f a, bb;
      a.x = Qs[lane][kk + 2 * half];
      a.y = Qs[lane][kk + 2 * half + 1];
      bb.x = K_[lane][kk + 2 * half];  // B[kd][n] = K[key n][kd]
      bb.y = K_[lane][kk + 2 * half + 1];
      accS = wmma4(a, bb, accS);
    }
    const bool masked = (kt * 16 + lane) >= kper;
#pragma unroll
    for (int j = 0; j < 8; ++j) {
      float s = accS[j] * scale;
      if (masked) s = -1e30f;
      Sp[j + 8 * half][kt * 16 + lane] = s;
    }
  }

  // ---- phase 2: row softmax over [0, nkt*16) ----
  {
    const int ncol = nkt * 16;
    const int row = lane;  // threads t and t+16 share a row, split cols
    float m = -1e30f;
    for (int c = half; c < ncol; c += 2) m = fmaxf(m, Sp[row][c]);
    m = fmaxf(m, __shfl_xor(m, 16, 32));
    float sum = 0.f;
    for (int c = half; c < ncol; c += 2) {
      const float e = __expf(Sp[row][c] - m);
      Sp[row][c] = e;
      sum += e;
    }
    sum += __shfl_xor(sum, 16, 32);
    const float inv = 1.f / sum;
    for (int c = half; c < ncol; c += 2) Sp[row][c] *= inv;
  }

  // ---- phase 3: O = P V ----
  v8f o0 = {0.f, 0.f, 0.f, 0.f, 0.f, 0.f, 0.f, 0.f};
  v8f o1 = {0.f, 0.f, 0.f, 0.f, 0.f, 0.f, 0.f, 0.f};
  stage_tile(Vs[0], qbase + 2 * C_, 16, ld, t);
  for (int kt = 0; kt < nkt; ++kt) {
    if (kt + 1 < nkt) {
      stage_tile(Vs[(kt + 1) & 1], qbase + 2 * C_ + (u64)((kt + 1) * 16) * ld,
                 16, ld, t);
      STAGE_WAIT(1);
    } else {
      STAGE_WAIT(0);
    }
    float(*V_)[36] = Vs[kt & 1];
#pragma unroll
    for (int kk = 0; kk < 16; kk += 4) {
      v2f a, b0, b1;
      a.x = Sp[lane][kt * 16 + kk + 2 * half];
      a.y = Sp[lane][kt * 16 + kk + 2 * half + 1];
      b0.x = V_[kk + 2 * half][lane];
      b0.y = V_[kk + 2 * half + 1][lane];
      b1.x = V_[kk + 2 * half][16 + lane];
      b1.y = V_[kk + 2 * half + 1][16 + lane];
      o0 = wmma4(a, b0, o0);
      o1 = wmma4(a, b1, o1);
    }
  }

#pragma unroll
  for (int j = 0; j < 8; ++j) {
    const int m = j + 8 * half;
    float* dst = ctx + (u64)(b * L_ + q0 + m) * C_ + h * D_;
    dst[lane] = o0[j];
    dst[16 + lane] = o1[j];
  }
}

// ---------------------------------------------------------------------------
// h = x * ratio[b]
// ---------------------------------------------------------------------------
__global__ __launch_bounds__(256) void resid_init_kernel(
    const float* __restrict__ x, const float* __restrict__ ratio,
    float* __restrict__ h) {
  const u64 i = (u64)blockIdx.x * 256 + threadIdx.x;
  const int row = (int)(i >> 8);  // / C_
  h[i] = x[i] * ratio[row >> 9];
}

// ---------------------------------------------------------------------------
// scatter: h[b, tidx[b,j], :] += h_attn[b,j,:] * ratio[b]   for j < kmax
// ---------------------------------------------------------------------------
__global__ __launch_bounds__(256) void scatter_kernel(
    const float* __restrict__ hattn, const int* __restrict__ tidx,
    const int* __restrict__ kmax_p, const float* __restrict__ ratio,
    float* __restrict__ h) {
  const int row = blockIdx.x;  // b*512 + j
  const int b = row >> 9, j = row & 511;
  if (j >= *kmax_p) return;
  const int pos = tidx[row];
  const int c = threadIdx.x;
  h[(u64)(b * L_ + pos) * C_ + c] += hattn[(u64)row * C_ + c] * ratio[b];
}

// ---------------------------------------------------------------------------
extern "C" void kernel_launch(void* const* d_in, const int* in_sizes, int n_in,
                              void* d_out, int out_size, void* d_ws,
                              size_t ws_size, hipStream_t stream) {
  const float* x = (const float*)d_in[0];
  const float* token_ratio = (const float*)d_in[1];
  const float* layer_gate = (const float*)d_in[2];
  const float* gumbel = (const float*)d_in[3];
  const float* scorer_w1 = (const float*)d_in[4];
  const float* scorer_b1 = (const float*)d_in[5];
  const float* scorer_w2 = (const float*)d_in[6];
  const float* scorer_b2 = (const float*)d_in[7];
  const float* in_proj_w = (const float*)d_in[8];
  const float* in_proj_b = (const float*)d_in[9];
  const float* out_proj_w = (const float*)d_in[10];
  const float* out_proj_b = (const float*)d_in[11];
  const float* mlp_w1 = (const float*)d_in[12];
  const float* mlp_b1 = (const float*)d_in[13];
  const float* mlp_w2 = (const float*)d_in[14];
  const float* mlp_b2 = (const float*)d_in[15];
  const int* kmax_p = (const int*)d_in[17];

  char* ws = (char*)d_ws;
  const size_t MB = (size_t)1 << 20;
  float* scores = (float*)(ws);                       // 64 KB
  int* tidx = (int*)(ws + 64 * 1024);                 // 64 KB
  int* kper = (int*)(ws + 128 * 1024);                // small
  float* hidden1 = (float*)(ws + 256 * 1024);         // 16 MB (later: ctx)
  float* hc = (float*)(ws + 256 * 1024 + 16 * MB);    // 16 MB (later: h_attn)
  float* qkv = (float*)(ws + 256 * 1024 + 32 * MB);   // 48 MB
  float* hbuf = (float*)(ws + 256 * 1024 + 80 * MB);  // 16 MB
  float* mlph = (float*)(ws + 256 * 1024 + 96 * MB);  // 32 MB
  float* ctx = hidden1;
  float* hattn = hc;

  wmma_gemm_kernel<<<dim3(N_ / 16, C_ / 32), 32, 0, stream>>>(
      x, scorer_w1, scorer_b1, hidden1, N_, C_, C_, 1, nullptr, nullptr,
      nullptr);
  score_reduce_kernel<<<N_, 32, 0, stream>>>(hidden1, scorer_w2, scorer_b2,
                                             scores);
  select_kernel<<<B_, L_, 0, stream>>>(scores, token_ratio, gumbel, kmax_p,
                                       tidx, kper);
  gather_kernel<<<B_ * L_, C_, 0, stream>>>(x, tidx, kper, hc);
  wmma_gemm_kernel<<<dim3(N_ / 16, (3 * C_) / 32), 32, 0, stream>>>(
      hc, in_proj_w, in_proj_b, qkv, N_, C_, 3 * C_, 0, nullptr, nullptr,
      nullptr);
  attn_kernel<<<dim3(L_ / 16, H_, B_), 32, 0, stream>>>(qkv, kper, kmax_p,
                                                        ctx);
  wmma_gemm_kernel<<<dim3(N_ / 16, C_ / 32), 32, 0, stream>>>(
      ctx, out_proj_w, out_proj_b, hattn, N_, C_, C_, 0, nullptr, nullptr,
      nullptr);
  resid_init_kernel<<<(N_ * C_) / 256, 256, 0, stream>>>(x, token_ratio, hbuf);
  scatter_kernel<<<B_ * L_, C_, 0, stream>>>(hattn, tidx, kmax_p, token_ratio,
                                             hbuf);
  wmma_gemm_kernel<<<dim3(N_ / 16, (2 * C_) / 32), 32, 0, stream>>>(
      hbuf, mlp_w1, mlp_b1, mlph, N_, C_, 2 * C_, 1, nullptr, nullptr,
      nullptr);
  wmma_gemm_kernel<<<dim3(N_ / 16, C_ / 32), 32, 0, stream>>>(
      mlph, mlp_w2, mlp_b2, (float*)d_out, N_, 2 * C_, C_, 2, hbuf, x,
      layer_gate);
}